// GNN_51049981280319
// MI455X (gfx1250) — compile-verified
//
#include <hip/hip_runtime.h>

typedef __attribute__((ext_vector_type(2))) float v2f;
typedef __attribute__((ext_vector_type(8))) float v8f;

// ---------------------------------------------------------------- utilities
__global__ void gnn_zero_kernel(float* __restrict__ p, int n) {
    int i = blockIdx.x * blockDim.x + threadIdx.x;
    int stride = gridDim.x * blockDim.x;
    for (; i < n; i += stride) p[i] = 0.0f;
}

// ------------------------------------------------- edge scatter (segment_sum)
// One wave per edge, each lane a float4 feature chunk: gather from x[src],
// atomic-add into agg[dst]. x and agg (25.6 MB each) are L2-resident (192 MB).
__global__ void gnn_scatter_kernel(const float* __restrict__ x,
                                   const long long* __restrict__ src,
                                   const long long* __restrict__ dst,
                                   float* __restrict__ agg, int E) {
    int g = blockIdx.x * blockDim.x + threadIdx.x;
    int e = g >> 5;
    int lane = g & 31;
    if (e >= E) return;
    int s = (int)src[e];
    int d = (int)dst[e];
    const float4 v = ((const float4*)(x + (size_t)s * 128))[lane];
    float* a = agg + (size_t)d * 128 + lane * 4;
    atomicAdd(a + 0, v.x);
    atomicAdd(a + 1, v.y);
    atomicAdd(a + 2, v.z);
    atomicAdd(a + 3, v.w);
}

// ------------------------------------- fused GraphConv GEMM via f32 WMMA
// Y = X @ Wroot + AGG @ Wrel + bias.
// Block = 8 waves, block tile = 64 rows x 128 cols. Wave w owns col tile
// [16w,16w+16) and 4 row sub-tiles; each B fragment is loaded once per K-step
// and reused for 4 WMMAs (4x B amortization).
__global__ void __launch_bounds__(256, 2)
gnn_gemm_fused_kernel(const float* __restrict__ X, const float* __restrict__ AGG,
                      const float* __restrict__ Wroot, const float* __restrict__ Wrel,
                      const float* __restrict__ bias, float* __restrict__ Y, int N) {
    constexpr int PITCH = 132;            // 128 + 4 pad: spreads the 16 lanes over distinct banks
    __shared__ float sX[64 * PITCH];
    __shared__ float sA[64 * PITCH];

    const int tid  = threadIdx.x;
    const int row0 = blockIdx.x * 64;

    // stage 64 rows of X and AGG (float4 coalesced, row-clamped)
    for (int i = tid; i < 64 * 32; i += 256) {          // 32 float4 per row
        int rl = i >> 5, c4 = (i & 31) * 4;
        int r  = row0 + rl;
        int rr = (r < N) ? r : (N - 1);
        const float4 vx = *(const float4*)(X   + (size_t)rr * 128 + c4);
        const float4 va = *(const float4*)(AGG + (size_t)rr * 128 + c4);
        *(float4*)(sX + rl * PITCH + c4) = vx;
        *(float4*)(sA + rl * PITCH + c4) = va;
    }
    __syncthreads();

    const int wave = tid >> 5;
    const int lane = tid & 31;
    const int half = lane >> 4;   // A/B frag: VGPR0 -> K = 2*half, VGPR1 -> K = 2*half+1
    const int mrow = lane & 15;
    const int col  = wave * 16 + mrow;

    v8f c0 = {}, c1 = {}, c2 = {}, c3 = {};

#pragma unroll
    for (int k = 0; k < 128; k += 4) {
        const int ka = k + 2 * half;
        v2f b;
        b.x = Wroot[ka * 128 + col];
        b.y = Wroot[(ka + 1) * 128 + col];
        v2f a0, a1, a2, a3;
        a0.x = sX[(mrow     ) * PITCH + ka]; a0.y = sX[(mrow     ) * PITCH + ka + 1];
        a1.x = sX[(mrow + 16) * PITCH + ka]; a1.y = sX[(mrow + 16) * PITCH + ka + 1];
        a2.x = sX[(mrow + 32) * PITCH + ka]; a2.y = sX[(mrow + 32) * PITCH + ka + 1];
        a3.x = sX[(mrow + 48) * PITCH + ka]; a3.y = sX[(mrow + 48) * PITCH + ka + 1];
        c0 = __builtin_amdgcn_wmma_f32_16x16x4_f32(false, a0, false, b, (short)0, c0, false, false);
        c1 = __builtin_amdgcn_wmma_f32_16x16x4_f32(false, a1, false, b, (short)0, c1, false, false);
        c2 = __builtin_amdgcn_wmma_f32_16x16x4_f32(false, a2, false, b, (short)0, c2, false, false);
        c3 = __builtin_amdgcn_wmma_f32_16x16x4_f32(false, a3, false, b, (short)0, c3, false, false);
    }
#pragma unroll
    for (int k = 0; k < 128; k += 4) {
        const int ka = k + 2 * half;
        v2f b;
        b.x = Wrel[ka * 128 + col];
        b.y = Wrel[(ka + 1) * 128 + col];
        v2f a0, a1, a2, a3;
        a0.x = sA[(mrow     ) * PITCH + ka]; a0.y = sA[(mrow     ) * PITCH + ka + 1];
        a1.x = sA[(mrow + 16) * PITCH + ka]; a1.y = sA[(mrow + 16) * PITCH + ka + 1];
        a2.x = sA[(mrow + 32) * PITCH + ka]; a2.y = sA[(mrow + 32) * PITCH + ka + 1];
        a3.x = sA[(mrow + 48) * PITCH + ka]; a3.y = sA[(mrow + 48) * PITCH + ka + 1];
        c0 = __builtin_amdgcn_wmma_f32_16x16x4_f32(false, a0, false, b, (short)0, c0, false, false);
        c1 = __builtin_amdgcn_wmma_f32_16x16x4_f32(false, a1, false, b, (short)0, c1, false, false);
        c2 = __builtin_amdgcn_wmma_f32_16x16x4_f32(false, a2, false, b, (short)0, c2, false, false);
        c3 = __builtin_amdgcn_wmma_f32_16x16x4_f32(false, a3, false, b, (short)0, c3, false, false);
    }

    const float bb = bias[col];
    const v8f acc[4] = {c0, c1, c2, c3};
#pragma unroll
    for (int m = 0; m < 4; ++m) {
#pragma unroll
        for (int v = 0; v < 8; ++v) {
            // C layout: VGPR v -> M = v (lanes 0-15), M = v+8 (lanes 16-31)
            int r = row0 + m * 16 + v + 8 * half;
            if (r < N) Y[(size_t)r * 128 + col] = acc[m][v] + bb;
        }
    }
}

// ------------------------------------------ batch-norm column statistics
// stats[0..127] = column sums, stats[128..255] = column sums of squares.
__global__ void gnn_stats_kernel(const float* __restrict__ Y, float* __restrict__ stats, int N) {
    __shared__ float ssum[256];
    __shared__ float ssq[256];
    const int tid = threadIdx.x;
    const int col = tid & 127;
    const int r0  = blockIdx.x * 2 + (tid >> 7);
    const int rs  = gridDim.x * 2;
    float s = 0.0f, q = 0.0f;
    for (int r = r0; r < N; r += rs) {
        float v = Y[(size_t)r * 128 + col];
        s += v;
        q += v * v;
    }
    ssum[tid] = s;
    ssq[tid]  = q;
    __syncthreads();
    if (tid < 128) {
        atomicAdd(&stats[col],       ssum[tid] + ssum[tid + 128]);
        atomicAdd(&stats[128 + col], ssq[tid]  + ssq[tid + 128]);
    }
}

// ------------------------------------------ apply batch norm (+optional ReLU)
__global__ void gnn_bn_relu_kernel(float* __restrict__ Y, const float* __restrict__ stats,
                                   const float* __restrict__ gamma, const float* __restrict__ beta,
                                   int N, int relu) {
    int i = blockIdx.x * blockDim.x + threadIdx.x;   // one float4 per thread
    if (i >= N * 32) return;
    float4* Y4 = (float4*)Y;
    float4 v = Y4[i];
    const int c0 = (i & 31) * 4;
    const float invN = 1.0f / (float)N;
    float* vp = &v.x;
#pragma unroll
    for (int j = 0; j < 4; ++j) {
        const int c = c0 + j;
        float mean = stats[c] * invN;
        float var  = stats[128 + c] * invN - mean * mean;
        float sc   = rsqrtf(var + 1e-5f) * gamma[c];
        float val  = (vp[j] - mean) * sc + beta[c];
        if (relu) val = fmaxf(val, 0.0f);
        vp[j] = val;
    }
    Y4[i] = v;
}

// ------------------------------------------ global mean pool (sums + counts)
__global__ void gnn_pool_kernel(const float* __restrict__ H,
                                const long long* __restrict__ batch,
                                float* __restrict__ pooled, float* __restrict__ counts, int N) {
    int g = blockIdx.x * blockDim.x + threadIdx.x;
    int n = g >> 5;
    int lane = g & 31;
    if (n >= N) return;
    int b = (int)batch[n];
    const float4 v = ((const float4*)(H + (size_t)n * 128))[lane];
    float* p = pooled + (size_t)b * 128 + lane * 4;
    atomicAdd(p + 0, v.x);
    atomicAdd(p + 1, v.y);
    atomicAdd(p + 2, v.z);
    atomicAdd(p + 3, v.w);
    if (lane == 0) atomicAdd(&counts[b], 1.0f);
}

// ------------------------------------------ classifier: (pooled/count) @ Wc + bc
__global__ void gnn_classify_kernel(const float* __restrict__ pooled,
                                    const float* __restrict__ counts,
                                    const float* __restrict__ Wc, const float* __restrict__ bc,
                                    float* __restrict__ out) {
    __shared__ float sP[128];
    const int g = blockIdx.x;
    const int t = threadIdx.x;
    const float cnt = fmaxf(counts[g], 1.0f);
    sP[t] = pooled[(size_t)g * 128 + t] / cnt;
    __syncthreads();
    if (t < 10) {
        float acc = bc[t];
#pragma unroll 16
        for (int k = 0; k < 128; ++k) acc += sP[k] * Wc[k * 10 + t];
        out[g * 10 + t] = acc;
    }
}

// ---------------------------------------------------------------- launcher
extern "C" void kernel_launch(void* const* d_in, const int* in_sizes, int n_in,
                              void* d_out, int out_size, void* d_ws, size_t ws_size,
                              hipStream_t stream) {
    const float*      x      = (const float*)d_in[0];
    const long long*  ei     = (const long long*)d_in[1];
    const long long*  batch  = (const long long*)d_in[2];
    const float* w_root[3]   = {(const float*)d_in[3], (const float*)d_in[6], (const float*)d_in[9]};
    const float* w_rel[3]    = {(const float*)d_in[4], (const float*)d_in[7], (const float*)d_in[10]};
    const float* bvec[3]     = {(const float*)d_in[5], (const float*)d_in[8], (const float*)d_in[11]};
    const float* gam[3]      = {(const float*)d_in[12], (const float*)d_in[14], (const float*)d_in[16]};
    const float* bet[3]      = {(const float*)d_in[13], (const float*)d_in[15], (const float*)d_in[17]};
    const float* w_cls       = (const float*)d_in[18];
    const float* b_cls       = (const float*)d_in[19];
    float* out = (float*)d_out;

    const int N = in_sizes[0] / 128;
    const int E = in_sizes[1] / 2;
    const int G = out_size / 10;
    const long long* src = ei;
    const long long* dst = ei + E;

    float* ws = (float*)d_ws;
    const size_t NH = (size_t)N * 128;
    float* agg    = ws;
    float* hA     = ws + NH;
    float* hB     = ws + 2 * NH;
    float* stats  = ws + 3 * NH;                 // 256 floats
    float* pooled = stats + 256;                 // G*128 floats
    float* counts = pooled + (size_t)G * 128;    // G floats

    const int scatterBlocks = (E * 32 + 255) / 256;
    const int gemmBlocks    = (N + 63) / 64;
    const int bnBlocks      = (N * 32 + 255) / 256;

    const float* in = x;
    float* y = hA;
    for (int l = 0; l < 3; ++l) {
        gnn_zero_kernel<<<4096, 256, 0, stream>>>(agg, (int)NH);
        gnn_scatter_kernel<<<scatterBlocks, 256, 0, stream>>>(in, src, dst, agg, E);
        gnn_gemm_fused_kernel<<<gemmBlocks, 256, 0, stream>>>(in, agg, w_root[l], w_rel[l],
                                                              bvec[l], y, N);
        gnn_zero_kernel<<<1, 256, 0, stream>>>(stats, 256);
        gnn_stats_kernel<<<128, 256, 0, stream>>>(y, stats, N);
        gnn_bn_relu_kernel<<<bnBlocks, 256, 0, stream>>>(y, stats, gam[l], bet[l], N,
                                                         (l < 2) ? 1 : 0);
        in = y;
        y  = (y == hA) ? hB : hA;
    }
    // `in` now points at the layer-3 output.
    gnn_zero_kernel<<<(G * 128 + 255) / 256, 256, 0, stream>>>(pooled, G * 128);
    gnn_zero_kernel<<<(G + 255) / 256, 256, 0, stream>>>(counts, G);
    gnn_pool_kernel<<<(N * 32 + 255) / 256, 256, 0, stream>>>(in, batch, pooled, counts, N);
    gnn_classify_kernel<<<G, 128, 0, stream>>>(pooled, counts, w_cls, b_cls, out);
}